// FocalLoss_22393959482076
// MI455X (gfx1250) — compile-verified
//
#include <hip/hip_runtime.h>
#include <hip/hip_bf16.h>

typedef __attribute__((ext_vector_type(2))) float v2f;
typedef __attribute__((ext_vector_type(8))) float v8f;

#define FL_C 80
#define FL_BLOCK 256
#define FL_GRID 2048

// ---------------------------------------------------------------------------
// Wave32 sum via V_WMMA_F32_16X16X4_F32:
//   A (16x4 f32, 2 VGPRs/lane): VGPR0 = acc, VGPR1 = 0
//     -> A[m][0] = acc[lane m], A[m][2] = acc[lane 16+m], A[*][1]=A[*][3]=0
//   B (4x16 f32) = all ones
//   D[m][n] = acc[m] + acc[16+m]  (= s[m], replicated across n)
//   D layout: VGPR j holds s[j] in lanes 0-15 and s[j+8] in lanes 16-31.
//   Sum 8 D-VGPRs per lane, then swap halves with one ds_swizzle (SWAPX16)
//   and add -> every lane holds the full 32-lane sum.
// ---------------------------------------------------------------------------
__device__ __forceinline__ float wave_sum_wmma(float acc) {
    v2f a;
    a[0] = acc;
    a[1] = 0.0f;
    v2f b;
    b[0] = 1.0f;
    b[1] = 1.0f;
    v8f c = {};
    c = __builtin_amdgcn_wmma_f32_16x16x4_f32(
        /*neg_a=*/false, a, /*neg_b=*/false, b,
        /*c_mod=*/(short)0, c, /*reuse_a=*/false, /*reuse_b=*/false);
    float s = ((c[0] + c[1]) + (c[2] + c[3])) + ((c[4] + c[5]) + (c[6] + c[7]));
    // group-of-32 swizzle: xor_mask=0x10, and_mask=0x1f -> offset 0x401F (SWAPX16)
    int sw = __builtin_amdgcn_ds_swizzle(__float_as_int(s), 0x401F);
    return s + __int_as_float(sw);
}

// focal element:  x = is_tgt ? v : -v
//   t     = exp(-|x|)
//   logpt = min(x,0) - log1p(t)          ( = log_sigmoid(x) )
//   1-pt  = x>=0 ? t/(1+t) : 1/(1+t)     ( = sigmoid(-x) )
//   loss  = -(1-pt)^2 * logpt
__device__ __forceinline__ float focal_elem(float v, bool is_tgt) {
    float x   = is_tgt ? v : -v;
    float ax  = fabsf(x);
    float t   = __expf(-ax);            // v_exp_f32 (scaled)
    float opt = 1.0f + t;
    float inv = __frcp_rn(opt);         // v_rcp_f32
    float L   = __logf(opt);            // v_log_f32 (scaled) = log1p(t)
    float lp  = fminf(x, 0.0f) - L;     // log_sigmoid(x) <= 0
    float omp = (x >= 0.0f) ? t * inv : inv;  // 1 - pt
    return -(omp * omp) * lp;
}

__global__ void __launch_bounds__(FL_BLOCK)
focal_partial_kernel(const float* __restrict__ inp,
                     const int* __restrict__ tgt32,   // int64 targets, low words
                     float* __restrict__ partial,
                     int nvec4, int total) {
    const unsigned tid    = threadIdx.x;
    const unsigned gid    = blockIdx.x * FL_BLOCK + tid;
    const unsigned stride = gridDim.x * FL_BLOCK;

    const float4* __restrict__ in4 = (const float4*)inp;

    float acc = 0.0f;
    for (unsigned j = gid; j < (unsigned)nvec4; j += stride) {
        // speculative prefetch one grid-stride ahead (OOB is silently dropped)
        __builtin_prefetch(&in4[j + stride], 0, 0);

        float4 v = in4[j];
        unsigned e   = j * 4u;             // flat element index (< 2^27, fits u32)
        unsigned row = e / (unsigned)FL_C; // mul-shift strength reduction
        unsigned col = e - row * (unsigned)FL_C;
        int t = tgt32[2u * row];           // low word of int64 target (0..79)

        int d = t - (int)col;              // which of the 4 elements is the target
        acc += focal_elem(v.x, d == 0);
        acc += focal_elem(v.y, d == 1);
        acc += focal_elem(v.z, d == 2);
        acc += focal_elem(v.w, d == 3);
    }

    // scalar tail (total not divisible by 4) -- handled by global thread 0
    if (gid == 0) {
        for (int e = nvec4 * 4; e < total; ++e) {
            unsigned row = (unsigned)e / (unsigned)FL_C;
            unsigned col = (unsigned)e - row * (unsigned)FL_C;
            acc += focal_elem(inp[e], (int)col == tgt32[2u * row]);
        }
    }

    // wave32 reduction on the matrix unit (EXEC all-1s here: fully reconverged)
    float ws = wave_sum_wmma(acc);

    __shared__ float sdata[FL_BLOCK / 32];
    const unsigned lane = tid & 31u;
    const unsigned wid  = tid >> 5;
    if (lane == 0) sdata[wid] = ws;
    __syncthreads();
    if (tid == 0) {
        float bs = 0.0f;
        #pragma unroll
        for (int w = 0; w < FL_BLOCK / 32; ++w) bs += sdata[w];
        partial[blockIdx.x] = bs;
    }
}

__global__ void __launch_bounds__(FL_BLOCK)
focal_final_kernel(const float* __restrict__ partial, int n,
                   float* __restrict__ out, float inv_n) {
    const unsigned tid = threadIdx.x;
    float acc = 0.0f;
    for (int i = (int)tid; i < n; i += FL_BLOCK) acc += partial[i];

    float ws = wave_sum_wmma(acc);

    __shared__ float sdata[FL_BLOCK / 32];
    const unsigned lane = tid & 31u;
    const unsigned wid  = tid >> 5;
    if (lane == 0) sdata[wid] = ws;
    __syncthreads();
    if (tid == 0) {
        float s = 0.0f;
        #pragma unroll
        for (int w = 0; w < FL_BLOCK / 32; ++w) s += sdata[w];
        out[0] = s * inv_n;   // mean over N rows
    }
}

extern "C" void kernel_launch(void* const* d_in, const int* in_sizes, int n_in,
                              void* d_out, int out_size, void* d_ws, size_t ws_size,
                              hipStream_t stream) {
    const float* inp   = (const float*)d_in[0];
    const int*   tgt32 = (const int*)d_in[1];   // int64 targets; read low words

    const int total = in_sizes[0];          // N * C = 83,886,080
    const int nrows = in_sizes[1];          // N = 1,048,576
    const int nvec4 = total / 4;

    int grid = FL_GRID;
    if (ws_size < (size_t)grid * sizeof(float))
        grid = (int)(ws_size / sizeof(float));   // defensive clamp, deterministic
    if (grid < 1) grid = 1;

    float* partials = (float*)d_ws;

    focal_partial_kernel<<<grid, FL_BLOCK, 0, stream>>>(inp, tgt32, partials,
                                                        nvec4, total);
    focal_final_kernel<<<1, FL_BLOCK, 0, stream>>>(partials, grid,
                                                   (float*)d_out,
                                                   1.0f / (float)nrows);
}